// SelfAttentionDot_32590211842688
// MI455X (gfx1250) — compile-verified
//
#include <hip/hip_runtime.h>
#include <cstdint>
#include <cstddef>

typedef __attribute__((ext_vector_type(2))) float v2f;
typedef __attribute__((ext_vector_type(8))) float v8f;

#define HIDDEN 1024
#define ATTN   256
#define NBATCH 32
#define NSEQ   2048
#define NROWS  (NBATCH * NSEQ)   // 65536 flattened (b,n) rows
#define WAVES  4                 // waves per block in align kernel
#define NSPLIT 16                // n-splits in pool kernel

// ---- CDNA5 async global->LDS helpers (inline asm; ASYNCcnt tracked) ----
__device__ __forceinline__ void async_ld16(uint32_t lds_byte_addr, const void* gaddr) {
  asm volatile("global_load_async_to_lds_b128 %0, %1, off"
               :: "v"(lds_byte_addr), "v"(gaddr) : "memory");
}
__device__ __forceinline__ void wait_async_le8() {   // prior chunk landed (in-order)
  asm volatile("s_wait_asynccnt 8" ::: "memory");
}
__device__ __forceinline__ void wait_async0() {
  asm volatile("s_wait_asynccnt 0" ::: "memory");
}

// ---------------- K1: fold weights: v[d] = sum_a W[a,d] * c[a] ----------------
__global__ void fold_kernel(const float* __restrict__ W, const float* __restrict__ cu,
                            float* __restrict__ v) {
  int d = blockIdx.x * blockDim.x + threadIdx.x;   // 0..1023
  float acc = 0.f;
  for (int a = 0; a < ATTN; ++a) acc += W[a * HIDDEN + d] * cu[a];
  v[d] = acc;
}

// Stage one 16x64 fp32 chunk (4 KB) of H rows into a wave-private LDS tile
// via 8 async b128 ops (each: 32 lanes x 16 B).
__device__ __forceinline__ void stage_chunk(const float* __restrict__ H, int rowBase, int kb,
                                            float (*buf)[68], int lane) {
  #pragma unroll
  for (int i = 0; i < 8; ++i) {
    int idx = i * 32 + lane;                       // 0..255 -> (row, 16B col-group)
    int row = idx >> 4;
    int cg  = idx & 15;
    const float* gp = H + (size_t)(rowBase + row) * HIDDEN + kb + cg * 4;
    uint32_t la = (uint32_t)(uintptr_t)&buf[row][cg * 4];
    async_ld16(la, gp);
  }
}

// ---------------- K2: align[row] = <H[row,:], v> via V_WMMA_F32_16X16X4_F32 ----------------
// Per wave: 16 rows. A = 16x4 fp32 tile of H (from LDS), B = v-chunk broadcast into all
// 16 columns (every column of C equals align for those rows). Double-buffered async
// global->LDS so the HBM stream never stalls on the WMMA consumer.
__global__ void __launch_bounds__(WAVES * 32)
align_kernel(const float* __restrict__ H, const float* __restrict__ v,
             float* __restrict__ alignOut) {
  __shared__ float vsh[HIDDEN];                    // folded vector, shared by block
  __shared__ float tile[2][WAVES][16][68];         // double-buffered per-wave 16x64 staging

  const int tid  = threadIdx.x;
  const int wv   = tid >> 5;
  const int lane = tid & 31;

  #pragma unroll
  for (int j = 0; j < 8; ++j) vsh[tid + 128 * j] = v[tid + 128 * j];
  __syncthreads();

  const int rowBase = blockIdx.x * (WAVES * 16) + wv * 16;  // 1024 blocks * 64 rows
  const int r    = lane & 15;                      // A-matrix M index for this lane
  const int koff = (lane >> 4) * 2;                // lanes 0-15: K0,K1; lanes 16-31: K2,K3

  v8f c = {};                                      // fp32 accumulator, 16x16

  stage_chunk(H, rowBase, 0, tile[0][wv], lane);   // prologue: chunk 0 in flight

  for (int ch = 0; ch < 16; ++ch) {                // 16 chunks of K=64 -> full D=1024
    if (ch + 1 < 16) {
      stage_chunk(H, rowBase, (ch + 1) * 64, tile[(ch + 1) & 1][wv], lane);
      wait_async_le8();                            // only chunk ch's 8 ops must be done
    } else {
      wait_async0();
    }
    const float (*cur)[68] = tile[ch & 1][wv];
    const int kb = ch * 64;
    #pragma unroll
    for (int s = 0; s < 16; ++s) {                 // 16 WMMA steps consume K=64
      const int k = s * 4;
      v2f a, bb;
      a.x  = cur[r][k + koff];
      a.y  = cur[r][k + koff + 1];
      bb.x = vsh[kb + k + koff];                   // broadcast v into all 16 B columns
      bb.y = vsh[kb + k + koff + 1];
      c = __builtin_amdgcn_wmma_f32_16x16x4_f32(false, a, false, bb,
                                                (short)0, c, false, false);
    }
  }

  // All 16 columns of C are identical; lane0 holds (M=0..7,N=0), lane16 holds (M=8..15,N=0)
  if (lane == 0) {
    #pragma unroll
    for (int m = 0; m < 8; ++m) alignOut[rowBase + m] = c[m];
  } else if (lane == 16) {
    #pragma unroll
    for (int m = 0; m < 8; ++m) alignOut[rowBase + 8 + m] = c[m];
  }
}

// ---------------- K3: softmax over n per batch (in place) + zero d_out ----------------
__global__ void softmax_kernel(float* __restrict__ buf, float* __restrict__ out) {
  __shared__ float red[256];
  const int b = blockIdx.x;
  const int tid = threadIdx.x;

  float vals[8];
  #pragma unroll
  for (int j = 0; j < 8; ++j) vals[j] = buf[b * NSEQ + tid + 256 * j];

  float m = vals[0];
  #pragma unroll
  for (int j = 1; j < 8; ++j) m = fmaxf(m, vals[j]);
  red[tid] = m; __syncthreads();
  for (int s = 128; s > 0; s >>= 1) {
    if (tid < s) red[tid] = fmaxf(red[tid], red[tid + s]);
    __syncthreads();
  }
  m = red[0]; __syncthreads();

  float e[8]; float sum = 0.f;
  #pragma unroll
  for (int j = 0; j < 8; ++j) { e[j] = expf(vals[j] - m); sum += e[j]; }
  red[tid] = sum; __syncthreads();
  for (int s = 128; s > 0; s >>= 1) {
    if (tid < s) red[tid] += red[tid + s];
    __syncthreads();
  }
  const float inv = 1.f / red[0];

  #pragma unroll
  for (int j = 0; j < 8; ++j) buf[b * NSEQ + tid + 256 * j] = e[j] * inv;

  // zero output for the atomic accumulation pass (harness poisons d_out)
  #pragma unroll
  for (int k = 0; k < 4; ++k) out[b * HIDDEN + tid + 256 * k] = 0.f;
}

// ---------------- K4: out[b,d] = sum_n attn[b,n] * H[b,n,d] ----------------
// Each lane owns 4 consecutive d (float4 -> global_load_b128, 4 KB/block/step coalesced);
// 16-way n-split for latency hiding, fp32 atomics to combine partials.
__global__ void pool_kernel(const float* __restrict__ H, const float* __restrict__ attn,
                            float* __restrict__ out) {
  const int b  = blockIdx.x;
  const int n0 = blockIdx.y * (NSEQ / NSPLIT);     // 128 rows per block
  const int d4 = threadIdx.x;                      // 0..255 -> d = 4*d4
  const float4* hb = (const float4*)(H + ((size_t)b * NSEQ + n0) * HIDDEN) + d4;
  const float*  ab = attn + b * NSEQ + n0;
  float4 acc = make_float4(0.f, 0.f, 0.f, 0.f);
  for (int n = 0; n < NSEQ / NSPLIT; ++n) {
    const float w = ab[n];
    const float4 h = hb[(size_t)n * (HIDDEN / 4)];
    acc.x += w * h.x; acc.y += w * h.y; acc.z += w * h.z; acc.w += w * h.w;
  }
  float* o = out + b * HIDDEN + d4 * 4;
  atomicAdd(o + 0, acc.x);
  atomicAdd(o + 1, acc.y);
  atomicAdd(o + 2, acc.z);
  atomicAdd(o + 3, acc.w);
}

extern "C" void kernel_launch(void* const* d_in, const int* in_sizes, int n_in,
                              void* d_out, int out_size, void* d_ws, size_t ws_size,
                              hipStream_t stream) {
  const float* H  = (const float*)d_in[0];   // [32, 2048, 1024]
  const float* W  = (const float*)d_in[1];   // [256, 1024]
  const float* cu = (const float*)d_in[2];   // [256, 1]
  float* out = (float*)d_out;                // [32, 1024]

  float* v    = (float*)d_ws;                // [1024]
  float* attn = v + HIDDEN;                  // [65536] align scores, then attn in place

  fold_kernel   <<<HIDDEN / 256, 256, 0, stream>>>(W, cu, v);
  align_kernel  <<<NROWS / (WAVES * 16), WAVES * 32, 0, stream>>>(H, v, attn);
  softmax_kernel<<<NBATCH, 256, 0, stream>>>(attn, out);
  pool_kernel   <<<dim3(NBATCH, NSPLIT), 256, 0, stream>>>(H, attn, out);
}